// Net4_90486370992283
// MI455X (gfx1250) — compile-verified
//
#include <hip/hip_runtime.h>

// ---------------------------------------------------------------------------
// GCN 4-layer forward for MI455X (gfx1250, wave32).
//
// Per layer: g = h @ W first (dense GEMM via V_WMMA_F32_16X16X4_F32, f32
// in / f32 accumulate), then graph aggregation agg[dst] += g[src] with
// global f32 atomics (edge scatter-add; destinations are L2-resident).
// GEMM-first shrinks aggregation width for layers 3 (128) and 4 (2).
// ReLU is folded into the A-operand load of the consuming GEMM (template
// parameter -> branch-free); bias is folded into the agg-buffer init.
// W tile is staged in LDS (pair-interleaved) so B fragments are single
// ds_load_b64 ops at immediate offsets.
//
// Workspace: needs 2 * N * 256 * 4 bytes (~205 MB) in d_ws.
// Assumes N_NODES is a multiple of 16 (100000 = 6250*16).
// ---------------------------------------------------------------------------

typedef float v2f __attribute__((ext_vector_type(2)));
typedef float v8f __attribute__((ext_vector_type(8)));

#define WAVES_PER_BLOCK 8   // 256 threads = 8 wave32
#define KMAX 256            // max reduction dim staged in LDS (64KB tile)

// ---------------------------------------------------------------------------
// G[M,Nout] = (RELU? max(H,0) : H)[M,K] @ W[K,Nout]
// One wave computes a 16x64 output tile: 4 x v8f accumulators, A reused.
//
// V_WMMA_F32_16X16X4_F32 operand layout (wave32):
//   A 16x4 : lane 0-15 -> M rows, VGPR r + laneHalf h -> K = r + 2h
//            => per lane a contiguous float2 at H[row, k + 2h].
//   B 4x16 : mirrored — lane 0-15 -> N cols, VGPR r + half h -> K = r + 2h
//            => b = (W[k+2h, n], W[k+2h+1, n]).
//   C/D    : VGPR j, lane half h -> row j + 8h, lane 0-15 -> col.
//
// LDS layout: wtile[(k>>1)*128 + c*2 + (k&1)] = W[k][n0+c], so the B
// fragment for (k, half h, col c) is the 8-byte pair at
// float-offset (k/2 + h)*128 + c*2  -> one ds_load_b64.
// ---------------------------------------------------------------------------
template <int RELU>
__global__ void gemm_wmma_f32(const float* __restrict__ H,
                              const float* __restrict__ W,
                              float* __restrict__ G,
                              int M, int K, int Nout) {
  __shared__ float wtile[KMAX * 64];   // 64 KB

  const int lane = threadIdx.x & 31;
  const int wave = threadIdx.x >> 5;
  const int half = lane >> 4;          // 0: lanes 0-15, 1: lanes 16-31
  const int lid  = lane & 15;
  const int n0   = blockIdx.y * 64;

  // Cooperative, coalesced stage of W[:, n0:n0+64] into LDS (pair-interleaved).
  for (int idx = threadIdx.x; idx < K * 64; idx += blockDim.x) {
    const int k = idx >> 6;
    const int c = idx & 63;
    wtile[(k >> 1) * 128 + c * 2 + (k & 1)] = W[(size_t)k * Nout + n0 + c];
  }
  __syncthreads();

  const int m0 = (blockIdx.x * WAVES_PER_BLOCK + wave) * 16;
  if (m0 >= M) return;                 // wave-uniform exit (after the barrier)

  v8f acc0 = {}, acc1 = {}, acc2 = {}, acc3 = {};

  const float* hptr = H + (size_t)(m0 + lid) * K + 2 * half;
  const float* wl   = &wtile[half * 128 + lid * 2];

  for (int k = 0; k < K; k += 4) {
    float ax = hptr[0];
    float ay = hptr[1];
    if (RELU) { ax = fmaxf(ax, 0.0f); ay = fmaxf(ay, 0.0f); }
    const v2f a = { ax, ay };

    const v2f b0 = *(const v2f*)(wl + 0);    // cols n0+ 0..15
    const v2f b1 = *(const v2f*)(wl + 32);   // cols n0+16..31
    const v2f b2 = *(const v2f*)(wl + 64);   // cols n0+32..47
    const v2f b3 = *(const v2f*)(wl + 96);   // cols n0+48..63

    acc0 = __builtin_amdgcn_wmma_f32_16x16x4_f32(false, a, false, b0, (short)0, acc0, false, false);
    acc1 = __builtin_amdgcn_wmma_f32_16x16x4_f32(false, a, false, b1, (short)0, acc1, false, false);
    acc2 = __builtin_amdgcn_wmma_f32_16x16x4_f32(false, a, false, b2, (short)0, acc2, false, false);
    acc3 = __builtin_amdgcn_wmma_f32_16x16x4_f32(false, a, false, b3, (short)0, acc3, false, false);

    hptr += 4;
    wl   += 256;                        // (k/2) advances by 2 pairs of 128 floats
  }

#pragma unroll
  for (int j = 0; j < 8; ++j) {
    const int row = m0 + j + 8 * half;
    float* o = G + (size_t)row * Nout + n0 + lid;
    o[0]  = acc0[j];
    o[16] = acc1[j];
    o[32] = acc2[j];
    o[48] = acc3[j];
  }
}

// ---------------------------------------------------------------------------
// agg[i] = bias[i & (d-1)]   (d is a power of two: 256 / 128 / 2)
// ---------------------------------------------------------------------------
__global__ void init_bias(float* __restrict__ agg, const float* __restrict__ b,
                          long long total, int dmask) {
  long long i = (long long)blockIdx.x * blockDim.x + threadIdx.x;
  if (i < total) agg[i] = b[(int)(i & dmask)];
}

// ---------------------------------------------------------------------------
// Edge scatter-add: agg[dst[e], :] += g[src[e], :]  for d in {128, 256}.
// Flat mapping: thread T -> (edge, 4-float chunk). chunkShift = log2(d/4),
// so each wave covers one (d=128) or half of one (d=256) edge row ->
// 128-bit coalesced gather loads + 4 f32 atomics (resolve in 192MB L2).
// ---------------------------------------------------------------------------
__global__ void scatter_add(const float* __restrict__ g,
                            const int* __restrict__ src,
                            const int* __restrict__ dst,
                            float* __restrict__ agg,
                            int d, int chunkShift, long long totalChunks) {
  long long T = (long long)blockIdx.x * blockDim.x + threadIdx.x;
  if (T >= totalChunks) return;
  const int e = (int)(T >> chunkShift);
  const int c = (int)(T & ((1 << chunkShift) - 1));
  const int s = src[e];
  const int t = dst[e];
  const float4 v = *(const float4*)(g + (size_t)s * d + c * 4);
  float* o = agg + (size_t)t * d + c * 4;
  atomicAdd(o + 0, v.x);
  atomicAdd(o + 1, v.y);
  atomicAdd(o + 2, v.z);
  atomicAdd(o + 3, v.w);
}

// ---------------------------------------------------------------------------
// Head GEMM: g4[n, 0:2] = relu(h3[n, :]) @ W4[128, 2]  (tiny: scalar VALU,
// W4 staged in LDS).
// ---------------------------------------------------------------------------
__global__ void head_gemm(const float* __restrict__ H,
                          const float* __restrict__ W4,
                          float* __restrict__ G, int N) {
  __shared__ float w[256];
  if (threadIdx.x < 256) w[threadIdx.x] = W4[threadIdx.x];
  __syncthreads();

  const int n = blockIdx.x * blockDim.x + threadIdx.x;
  if (n >= N) return;
  const float* row = H + (size_t)n * 128;
  float a0 = 0.0f, a1 = 0.0f;
#pragma unroll 4
  for (int k = 0; k < 128; k += 4) {
    float4 v = *(const float4*)(row + k);
    v.x = fmaxf(v.x, 0.0f); v.y = fmaxf(v.y, 0.0f);
    v.z = fmaxf(v.z, 0.0f); v.w = fmaxf(v.w, 0.0f);
    a0 += v.x * w[2 * k + 0] + v.y * w[2 * k + 2] + v.z * w[2 * k + 4] + v.w * w[2 * k + 6];
    a1 += v.x * w[2 * k + 1] + v.y * w[2 * k + 3] + v.z * w[2 * k + 5] + v.w * w[2 * k + 7];
  }
  G[2 * n + 0] = a0;
  G[2 * n + 1] = a1;
}

// ---------------------------------------------------------------------------
// Final scatter for d = 2: one thread per edge, float2 load + 2 atomics.
// ---------------------------------------------------------------------------
__global__ void scatter_add2(const float* __restrict__ g,
                             const int* __restrict__ src,
                             const int* __restrict__ dst,
                             float* __restrict__ out, int E) {
  const int e = blockIdx.x * blockDim.x + threadIdx.x;
  if (e >= E) return;
  const int s = src[e];
  const int t = dst[e];
  const float2 v = *(const float2*)(g + (size_t)s * 2);
  atomicAdd(out + (size_t)t * 2 + 0, v.x);
  atomicAdd(out + (size_t)t * 2 + 1, v.y);
}

// ---------------------------------------------------------------------------
// Launch
// ---------------------------------------------------------------------------
extern "C" void kernel_launch(void* const* d_in, const int* in_sizes, int n_in,
                              void* d_out, int out_size, void* d_ws, size_t ws_size,
                              hipStream_t stream) {
  const int D_IN = 256, H2 = 256, H3 = 256, H4 = 128, OUT = 2;

  const float* x   = (const float*)d_in[0];
  const int*   src = (const int*)  d_in[1];
  const int*   dst = (const int*)  d_in[2];
  const float* W1  = (const float*)d_in[3];
  const float* b1  = (const float*)d_in[4];
  const float* W2  = (const float*)d_in[5];
  const float* b2  = (const float*)d_in[6];
  const float* W3  = (const float*)d_in[7];
  const float* b3  = (const float*)d_in[8];
  const float* W4  = (const float*)d_in[9];
  const float* b4  = (const float*)d_in[10];
  float* out = (float*)d_out;

  const int N = in_sizes[0] / D_IN;   // 100000 (multiple of 16)
  const int E = in_sizes[1];          // 3200000

  // Workspace: two N x 256 f32 ping-pong buffers.
  float* P = (float*)d_ws;
  float* Q = P + (size_t)N * 256;

  const int strips = N / 16;
  const dim3 blk(256);
  const dim3 gemmGrid256((strips + WAVES_PER_BLOCK - 1) / WAVES_PER_BLOCK, 256 / 64);
  const dim3 gemmGrid128((strips + WAVES_PER_BLOCK - 1) / WAVES_PER_BLOCK, 128 / 64);

  auto blocks1d = [](long long total) { return (int)((total + 255) / 256); };

  const long long nd256 = (long long)N * 256;
  const long long nd128 = (long long)N * 128;
  const long long ec256 = (long long)E * (256 / 4); // chunkShift 6
  const long long ec128 = (long long)E * (128 / 4); // chunkShift 5

  // ---- Layer 1: g1 = x @ W1 -> P ; Q = b1 ; Q += A-scatter(P) ----
  gemm_wmma_f32<0><<<gemmGrid256, blk, 0, stream>>>(x, W1, P, N, D_IN, H2);
  init_bias<<<blocks1d(nd256), blk, 0, stream>>>(Q, b1, nd256, 255);
  scatter_add<<<blocks1d(ec256), blk, 0, stream>>>(P, src, dst, Q, 256, 6, ec256);

  // ---- Layer 2: g2 = relu(Q) @ W2 -> P ; Q = b2 ; Q += scatter(P) ----
  gemm_wmma_f32<1><<<gemmGrid256, blk, 0, stream>>>(Q, W2, P, N, H2, H3);
  init_bias<<<blocks1d(nd256), blk, 0, stream>>>(Q, b2, nd256, 255);
  scatter_add<<<blocks1d(ec256), blk, 0, stream>>>(P, src, dst, Q, 256, 6, ec256);

  // ---- Layer 3: g3 = relu(Q) @ W3 -> P (N x 128) ; Q = b3 ; scatter ----
  gemm_wmma_f32<1><<<gemmGrid128, blk, 0, stream>>>(Q, W3, P, N, H3, H4);
  init_bias<<<blocks1d(nd128), blk, 0, stream>>>(Q, b3, nd128, 127);
  scatter_add<<<blocks1d(ec128), blk, 0, stream>>>(P, src, dst, Q, 128, 5, ec128);

  // ---- Layer 4: g4 = relu(Q) @ W4 -> P (N x 2) ; out = b4 ; scatter ----
  head_gemm<<<blocks1d(N), blk, 0, stream>>>(Q, W4, P, N);
  init_bias<<<blocks1d((long long)N * OUT), blk, 0, stream>>>(out, b4, (long long)N * OUT, 1);
  scatter_add2<<<blocks1d(E), blk, 0, stream>>>(P, src, dst, out, E);
}